// Coefficients_15960098472232
// MI455X (gfx1250) — compile-verified
//
#include <hip/hip_runtime.h>
#include <cstdint>
#include <cstddef>

typedef float v4f __attribute__((ext_vector_type(4)));

#define SYS_DT 1e-6f

// ---------------------------------------------------------------------------
// Kernel 1: rows [0, N) :  [ M | 0 | 0 ]
// 2D grid: y = row, x = float4 column. One b128 NT store per thread.
// ---------------------------------------------------------------------------
__global__ void k_top(const float* __restrict__ M, float* __restrict__ out,
                      int N, int E, int C) {
    const int r  = blockIdx.y;
    const int c4 = blockIdx.x * blockDim.x + threadIdx.x;
    if (c4 >= (C >> 2)) return;
    const int c = c4 << 2;
    v4f v;
    if (c < E) {  // E multiple of 4 -> float4 entirely inside M
        v = *(const v4f*)(M + (size_t)r * E + c);
    } else {
        v = (v4f){0.f, 0.f, 0.f, 0.f};
    }
    __builtin_nontemporal_store(v, (v4f*)(out + (size_t)r * C + c));
}

// ---------------------------------------------------------------------------
// Kernel 2: rows [N, N+E), cols [0, 2E) :  [ 0 | I_E ]
// ---------------------------------------------------------------------------
__global__ void k_mid(float* __restrict__ out, int N, int E, int C) {
    const int i  = blockIdx.y;                       // element row
    const int c4 = blockIdx.x * blockDim.x + threadIdx.x;
    if (c4 >= (E >> 1)) return;                      // 2E/4 float4s
    const int c = c4 << 2;
    v4f v = (v4f){0.f, 0.f, 0.f, 0.f};
    const int d = E + i;                             // identity diagonal column
    if (d >= c && d < c + 4) v[d - c] = 1.0f;
    __builtin_nontemporal_store(v, (v4f*)(out + (size_t)(N + i) * C + c));
}

// ---------------------------------------------------------------------------
// Kernel 3: rows [N, N+E), cols [2E, 2E+N) :  -M^T
// 32x32 LDS tile, 32x8 threads (8 wave32). Tile fill = one CDNA5 async
// global->LDS b128 per thread (ASYNCcnt tracked), then conflict-free
// transposed b32 reads (pad 33) and coalesced NT stores.
// ---------------------------------------------------------------------------
#ifndef USE_ASYNC_LDS
#define USE_ASYNC_LDS 1
#endif

__global__ __launch_bounds__(256)
void k_tr(const float* __restrict__ M, float* __restrict__ out,
          int N, int E, int C) {
    __shared__ float tile[32][33];          // +1 pad: conflict-free transposed reads
    const int i0 = blockIdx.x * 32;         // column block of M == row block of out
    const int j0 = blockIdx.y * 32;         // row block of M    == col block of out
    const int tx = threadIdx.x;             // 0..31
    const int ty = threadIdx.y;             // 0..7
    const int tid = ty * 32 + tx;           // 0..255

    // Each thread moves 16 B: tile row j = tid/8, cols 4*(tid%8)..+3
    const int j  = tid >> 3;
    const int cc = (tid & 7) << 2;
#if USE_ASYNC_LDS
    {
        const float* gp = M + (size_t)(j0 + j) * E + (i0 + cc);
        // low 32 bits of a flat pointer into LDS == wave-relative LDS byte offset
        uint32_t lds_off = (uint32_t)(uintptr_t)&tile[j][cc];
        asm volatile("global_load_async_to_lds_b128 %0, %1, off"
                     :: "v"(lds_off), "v"(gp) : "memory");
    }
    asm volatile("s_wait_asynccnt 0" ::: "memory");
#else
    *(v4f*)&tile[j][cc] = *(const v4f*)(M + (size_t)(j0 + j) * E + (i0 + cc));
#endif
    __syncthreads();

    #pragma unroll
    for (int k = 0; k < 4; ++k) {
        const int i = ty + 8 * k;           // local out-row (column of M)
        const float val = -tile[tx][i];     // M[j0+tx][i0+i]
        __builtin_nontemporal_store(
            val, out + (size_t)(N + i0 + i) * C + (2 * E + j0 + tx));
    }
}

// ---------------------------------------------------------------------------
// Kernel 4: rows [N+E, N+2E) :  [ diag(z) | diag(y) | 0 ]
// sigmoid(x) > 0.5  <=>  x > 0 (no transcendental needed).
// ---------------------------------------------------------------------------
__global__ void k_elem(const float* __restrict__ params,
                       const float* __restrict__ sw,
                       const int* __restrict__ kinds,
                       const int* __restrict__ timep,
                       float* __restrict__ out,
                       int N, int E, int SIG, int C) {
    const int i  = blockIdx.y;                       // element row
    const int c4 = blockIdx.x * blockDim.x + threadIdx.x;
    if (c4 >= (C >> 2)) return;
    const int c = c4 << 2;

    v4f v = (v4f){0.f, 0.f, 0.f, 0.f};
    const int dz = i;        // column of z on first diagonal
    const int dy = E + i;    // column of y on second diagonal
    const bool hit_z = (dz >= c && dz < c + 4);
    const bool hit_y = (dy >= c && dy < c + 4);
    if (hit_z || hit_y) {
        const float p = params[i];
        const int kind = kinds[i];
        const int t = timep[0];
        const bool sw_on = sw[(size_t)i * SIG + t] > 0.0f;  // sigmoid(x) > 0.5
        const float ndt = -SYS_DT / p;
        float z, y;
        switch (kind) {
            case 0:  z = -p;                y = 1.0f;               break; // R
            case 1:  z = 0.0f;              y = 1.0f;               break; // IVS
            case 2:  z = 1.0f;              y = 0.0f;               break; // VC
            case 3:  z = sw_on ? 0.f : 1.f; y = sw_on ? 1.f : 0.f;  break; // SW
            case 4:  z = ndt;               y = 1.0f;               break; // C
            case 5:  z = 1.0f;              y = ndt;                break; // L
            default: z = 0.0f;              y = 0.0f;               break;
        }
        if (hit_z) v[dz - c] = z;
        if (hit_y) v[dy - c] = y;
    }
    __builtin_nontemporal_store(v, (v4f*)(out + (size_t)(N + E + i) * C + c));
}

// ---------------------------------------------------------------------------
extern "C" void kernel_launch(void* const* d_in, const int* in_sizes, int n_in,
                              void* d_out, int out_size, void* d_ws, size_t ws_size,
                              hipStream_t stream) {
    const float* M      = (const float*)d_in[0];
    const float* params = (const float*)d_in[1];
    const float* sw     = (const float*)d_in[2];
    const int*   kinds  = (const int*)d_in[3];
    const int*   timep  = (const int*)d_in[4];
    float* out = (float*)d_out;

    const int E   = in_sizes[1];          // 2048
    const int N   = in_sizes[0] / E;      // 1024
    const int SIG = in_sizes[2] / E;      // 64
    const int C   = 2 * E + N;            // 5120 columns (and 5120 rows total)

    const int c4n   = C >> 2;             // 1280 float4s per full row
    const int c4n2E = E >> 1;             // 2E/4 = 1024 float4s

    {   // rows [0,N): [M | 0]
        dim3 grid((c4n + 255) / 256, N);
        k_top<<<grid, 256, 0, stream>>>(M, out, N, E, C);
    }
    {   // rows [N,N+E), cols [0,2E): [0 | I]
        dim3 grid((c4n2E + 255) / 256, E);
        k_mid<<<grid, 256, 0, stream>>>(out, N, E, C);
    }
    {   // rows [N,N+E), cols [2E,C): -M^T via async-LDS tiled transpose
        dim3 grid(E / 32, N / 32);
        k_tr<<<grid, dim3(32, 8), 0, stream>>>(M, out, N, E, C);
    }
    {   // rows [N+E,N+2E): [diag(z) | diag(y) | 0]
        dim3 grid((c4n + 255) / 256, E);
        k_elem<<<grid, 256, 0, stream>>>(params, sw, kinds, timep, out, N, E, SIG, C);
    }
}